// SoftFireCA_wind_1374389534750
// MI455X (gfx1250) — compile-verified
//
#include <hip/hip_runtime.h>
#include <cstdint>
#include <cstddef>

#define HDIM 1024
#define WDIM 1024
#define NCELL (HDIM * WDIM)
#define NSTEPS 20
#define TILE 32
#define PITCH 34            // TILE + 2 halo
#define LDS_ELEMS 1280      // 5 * 256 >= 34*34 = 1156 (padded so every lane always loads)

__device__ __forceinline__ int iclamp(int x, int lo, int hi) {
    return x < lo ? lo : (x > hi ? hi : x);
}

// ---------------------------------------------------------------------------
// CDNA5 async global -> LDS copy (tracked by ASYNCcnt). Inline asm is the
// portable path across ROCm 7.2 / amdgpu-toolchain per the bridge doc.
// vdst operand = per-lane LDS byte offset (flat->local is a truncation),
// vaddr = per-lane 64-bit global address.
// ---------------------------------------------------------------------------
__device__ __forceinline__ void async_g2l_b32(const float* gptr, float* lptr) {
    unsigned int laddr = (unsigned int)(size_t)lptr;
    unsigned long long ga = (unsigned long long)(size_t)gptr;
    asm volatile("global_load_async_to_lds_b32 %0, %1, off"
                 :: "v"(laddr), "v"(ga)
                 : "memory");
}

__device__ __forceinline__ void wait_async0() {
    asm volatile("s_wait_asynccnt 0x0" ::: "memory");
}

// ---------------------------------------------------------------------------
// Pass 1: state-independent precompute.
//   gain[c]            = age_factor * exp(-beta * moisture)
//   coef[c*8 + k]      = dist_k * wind_factor_k * slope_factor_k * mask_k
//   state0[c]          = ignition impulse
//   arrival[c] (d_out) = n_steps (20)
// ---------------------------------------------------------------------------
__global__ __launch_bounds__(256) void precompute_kernel(
    const float* __restrict__ height, const float* __restrict__ age,
    const float* __restrict__ moist,
    const float* __restrict__ p_la, const float* __restrict__ p_lb,
    const float* __restrict__ p_lg, const float* __restrict__ p_ld,
    const float* __restrict__ wspd, const float* __restrict__ wdir,
    const float* __restrict__ p_igv, const int* __restrict__ p_ip,
    float* __restrict__ coef, float* __restrict__ gain,
    float* __restrict__ state0, float* __restrict__ arrival)
{
    const int cell = blockIdx.x * blockDim.x + threadIdx.x;
    if (cell >= NCELL) return;
    const int i = cell >> 10;           // WDIM == 1024
    const int j = cell & (WDIM - 1);

    // Uniform scalar parameters (lowered to s_load).
    const float alpha = expf(p_la[0]);
    const float beta  = expf(p_lb[0]);
    const float gamma = expf(p_lg[0]);
    const float delta = expf(p_ld[0]);

    // Peterson age inflammability: (1 + P_MAX)^(ratio^alpha) - 1, P_MAX = 1.
    const float a = age[cell];
    const float ratio = fmaxf(a * (1.0f / 30.0f), 1e-6f);
    float af = exp2f(expf(alpha * logf(ratio))) - 1.0f;   // 2^(ratio^alpha) - 1
    if (a >= 30.0f) af = 1.0f;                            // saturate at P_MAX
    gain[cell] = af * expf(-beta * moist[cell]);

    const float R = 0.70710678118654752f;
    const int   di[8]   = {-1, -1, -1,  0,  0,  1,  1,  1};
    const int   dj[8]   = {-1,  0,  1, -1,  1, -1,  0,  1};
    const float dist[8] = {0.83f, 1.0f, 0.83f, 1.0f, 1.0f, 0.83f, 1.0f, 0.83f};
    const float ui[8]   = {-R, -1.0f, -R, 0.0f, 0.0f,  R, 1.0f,  R};
    const float uj[8]   = {-R,  0.0f,  R, -1.0f, 1.0f, -R, 0.0f,  R};

    const float hc = height[cell];
    float c8[8];
#pragma unroll
    for (int k = 0; k < 8; ++k) {
        const int ii = i + di[k], jj = j + dj[k];
        const int ni = iclamp(ii, 0, HDIM - 1);
        const int nj = iclamp(jj, 0, WDIM - 1);
        const int nidx = ni * WDIM + nj;

        const float dh  = hc - height[nidx];
        const float phi = (dh <= 0.0f) ? expf(gamma * dh)
                                       : (1.0f + gamma * sqrtf(dh));
        float s, c;
        sincosf(wdir[nidx], &s, &c);
        const float wsn   = wspd[nidx];
        const float align = ui[k] * (wsn * c) + uj[k] * (wsn * s);   // ui*wind_y + uj*wind_x
        const float wf    = fminf(expf(delta * align), 2.0f);        // clip(exp,-2,2), exp>=0

        const bool ok = ((unsigned)ii < (unsigned)HDIM) && ((unsigned)jj < (unsigned)WDIM);
        c8[k] = ok ? (dist[k] * wf * phi) : 0.0f;
    }
    float4* cp = (float4*)(coef + (size_t)cell * 8u);
    cp[0] = make_float4(c8[0], c8[1], c8[2], c8[3]);
    cp[1] = make_float4(c8[4], c8[5], c8[6], c8[7]);

    state0[cell]  = (i == p_ip[0] && j == p_ip[1]) ? p_igv[0] : 0.0f;
    arrival[cell] = (float)NSTEPS;
}

// ---------------------------------------------------------------------------
// Pass 2 (x20): one CA step. 32x32 tile + halo staged into LDS with the
// CDNA5 async global->LDS path; each of 256 threads computes 4 cells.
// ---------------------------------------------------------------------------
__global__ __launch_bounds__(256) void fire_step_kernel(
    const float* __restrict__ src, float* __restrict__ dst,
    const float* __restrict__ coef, const float* __restrict__ gain,
    float* __restrict__ arrival, float wt)
{
    __shared__ float tile[LDS_ELEMS];
    const int tx  = threadIdx.x;            // 0..31
    const int ty  = threadIdx.y;            // 0..7
    const int bi  = blockIdx.y * TILE;
    const int bj  = blockIdx.x * TILE;
    const int tid = ty * 32 + tx;

    // Stage (TILE+2)^2 halo tile (edge-clamped; clamped cells carry zero coef).
#pragma unroll
    for (int k = 0; k < 5; ++k) {
        const int l  = tid + k * 256;       // 0..1279, all lanes always active
        const int li = l / PITCH;
        const int lj = l - li * PITCH;
        const int gi = iclamp(bi + li - 1, 0, HDIM - 1);
        const int gj = iclamp(bj + lj - 1, 0, WDIM - 1);
        async_g2l_b32(src + gi * WDIM + gj, &tile[l]);
    }
    wait_async0();
    __syncthreads();

#pragma unroll
    for (int rr = 0; rr < 4; ++rr) {
        const int r    = ty + rr * 8;       // 0..31
        const int ci   = bi + r;
        const int cj   = bj + tx;
        const int cell = ci * WDIM + cj;

        const float4* cp = (const float4*)(coef + (size_t)cell * 8u);
        const float4 c0 = cp[0];
        const float4 c1 = cp[1];

        const int lt = (r + 1) * PITCH + (tx + 1);
        const float sc = tile[lt];
        float tot;
        tot =       c0.x * tile[lt - PITCH - 1];
        tot = fmaf(c0.y, tile[lt - PITCH    ], tot);
        tot = fmaf(c0.z, tile[lt - PITCH + 1], tot);
        tot = fmaf(c0.w, tile[lt - 1        ], tot);
        tot = fmaf(c1.x, tile[lt + 1        ], tot);
        tot = fmaf(c1.y, tile[lt + PITCH - 1], tot);
        tot = fmaf(c1.z, tile[lt + PITCH    ], tot);
        tot = fmaf(c1.w, tile[lt + PITCH + 1], tot);

        const float sn = fminf(fmaxf(fmaf(gain[cell], tot, sc), 0.0f), 1.0f);
        dst[cell] = sn;

        const float fi = fmaxf(sn - sc, 0.0f);   // first ignition this step
        arrival[cell] -= fi * wt;                // wt = n_steps - t
    }
}

// ---------------------------------------------------------------------------
// Host launcher. Inputs (dict order): height, age, moisture, log_alpha,
// log_beta, log_gamma, log_delta, wind_speed, wind_dir, ignition_value,
// ignition_point(int2), n_steps, n_substeps. Reference fixes n_steps=20,
// n_substeps=1, H=W=1024.
// ---------------------------------------------------------------------------
extern "C" void kernel_launch(void* const* d_in, const int* in_sizes, int n_in,
                              void* d_out, int out_size, void* d_ws, size_t ws_size,
                              hipStream_t stream)
{
    (void)in_sizes; (void)n_in; (void)out_size;

    const float* height = (const float*)d_in[0];
    const float* age    = (const float*)d_in[1];
    const float* moist  = (const float*)d_in[2];
    const float* p_la   = (const float*)d_in[3];
    const float* p_lb   = (const float*)d_in[4];
    const float* p_lg   = (const float*)d_in[5];
    const float* p_ld   = (const float*)d_in[6];
    const float* wspd   = (const float*)d_in[7];
    const float* wdir   = (const float*)d_in[8];
    const float* p_igv  = (const float*)d_in[9];
    const int*   p_ip   = (const int*)d_in[10];

    // Workspace layout: coef[8*N] | gain[N] | stateA[N] | stateB[N]
    float* coef = (float*)d_ws;
    float* gain = coef + (size_t)8 * NCELL;
    float* sA   = gain + NCELL;
    float* sB   = sA + NCELL;
    float* arrival = (float*)d_out;
    if (ws_size < (size_t)11 * NCELL * sizeof(float)) return;

    precompute_kernel<<<NCELL / 256, 256, 0, stream>>>(
        height, age, moist, p_la, p_lb, p_lg, p_ld, wspd, wdir,
        p_igv, p_ip, coef, gain, sA, arrival);

    dim3 grd(WDIM / TILE, HDIM / TILE);
    dim3 blk(32, 8);
    float* s = sA;
    float* d = sB;
    for (int t = 1; t <= NSTEPS; ++t) {
        fire_step_kernel<<<grd, blk, 0, stream>>>(s, d, coef, gain, arrival,
                                                  (float)(NSTEPS - t));
        float* tmp = s; s = d; d = tmp;
    }
}